// ModifiedGCN_8177617732167
// MI455X (gfx1250) — compile-verified
//
#include <hip/hip_runtime.h>
#include <hip/hip_bf16.h>

// ---------------------------------------------------------------------------
// ModifiedGCN on MI455X (gfx1250): WMMA f32 GEMMs + atomic scatter GCN layers
// ---------------------------------------------------------------------------

#define GCN_N   100000
#define GCN_E   1250000
#define GCN_H   64
#define GCN_CIN 64
#define GCN_COUT 16
#define GCN_L   3
#define GCN_G   512
#define GCN_EPS 1e-5f

typedef __attribute__((ext_vector_type(2))) float v2f;
typedef __attribute__((ext_vector_type(8))) float v8f;

// ---------------------------------------------------------------------------
// GEMM: C[M,NOUT] = A[M,64] * B[64,NOUT] (+bias) (optional sigmoid)
// One wave32 per (16*MT)x16 output stripe. K=64 via 16x v_wmma_f32_16x16x4_f32
// per 16x16 tile. B is staged into 32 VGPRs once and reused across MT tiles.
//
// f32 A-operand VGPR layout (ISA 7.12.2, 16x4):
//   lanes 0-15:  M=lane,    VGPR0=K0,  VGPR1=K1
//   lanes 16-31: M=lane-16, VGPR0=K2,  VGPR1=K3
// B-operand (4x16) mirrors with columns striped across lanes.
// C/D: 8 VGPRs, element (M = r + 8*(lane>>4), N = lane&15).
// NOUT is a compile-time constant so all addresses fold to base+immediate.
// ---------------------------------------------------------------------------
template <int NOUT, int MT>
__global__ __launch_bounds__(32) void gcn_gemm64_wmma(
    const float* __restrict__ A,
    const float* __restrict__ B,
    const float* __restrict__ bias,   // may be nullptr
    float* __restrict__ C,
    int do_sigmoid)
{
    const int m0   = blockIdx.x * (16 * MT);
    const int n0   = blockIdx.y << 4;
    const int lane = threadIdx.x & 31;
    const int half = lane >> 4;       // 0 or 1
    const int l16  = lane & 15;
    const int bcol = n0 + l16;

    // ---- stage B operand: lane holds rows (4t+2h, 4t+2h+1), col = bcol ----
    v2f breg[16];
    {
        const float* __restrict__ bp = B + (2 * half) * NOUT + bcol;
#pragma unroll
        for (int t = 0; t < 16; ++t) {
            breg[t].x = bp[(4 * t) * NOUT];
            breg[t].y = bp[(4 * t + 1) * NOUT];
        }
    }

    const float bv = bias ? bias[bcol] : 0.0f;

#pragma unroll
    for (int mt = 0; mt < MT; ++mt) {
        const int mbase = m0 + mt * 16;
        const float* __restrict__ ap =
            A + (long long)(mbase + l16) * 64 + 2 * half;
        v8f c = {};
#pragma unroll
        for (int t = 0; t < 16; ++t) {
            v2f a;
            a.x = ap[4 * t];
            a.y = ap[4 * t + 1];
            c = __builtin_amdgcn_wmma_f32_16x16x4_f32(
                    /*neg_a=*/false, a, /*neg_b=*/false, breg[t],
                    /*c_mod=*/(short)0, c, /*reuse_a=*/false, /*reuse_b=*/false);
        }
#pragma unroll
        for (int r = 0; r < 8; ++r) {
            const int m = mbase + r + 8 * half;
            float v = c[r] + bv;
            if (do_sigmoid) v = 1.0f / (1.0f + expf(-v));
            C[(long long)m * NOUT + bcol] = v;
        }
    }
}

// ---------------------------------------------------------------------------
// Elementwise / scatter helpers (float4-vectorized: 4 channels per thread)
// ---------------------------------------------------------------------------
__global__ void gcn_fill(float* __restrict__ p, long long n, float v)
{
    long long i = (long long)blockIdx.x * blockDim.x + threadIdx.x;
    const long long s = (long long)gridDim.x * blockDim.x;
    for (; i < n; i += s) p[i] = v;
}

__global__ void gcn_count_accum(const int* __restrict__ idx,
                                float* __restrict__ acc, int n)
{
    const int i = blockIdx.x * blockDim.x + threadIdx.x;
    if (i < n) atomicAdd(&acc[idx[i]], 1.0f);
}

__global__ void gcn_rsqrt_inplace(float* __restrict__ p, int n)
{
    const int i = blockIdx.x * blockDim.x + threadIdx.x;
    if (i < n) p[i] = rsqrtf(p[i]);
}

// agg[i, c4..c4+3] = hw * dinv[i]^2 (self-loop) + convb ; total = N*16
__global__ void gcn_agg_init(const float* __restrict__ hw,
                             const float* __restrict__ dinv,
                             const float* __restrict__ convb,
                             float* __restrict__ agg, long long total)
{
    long long idx = (long long)blockIdx.x * blockDim.x + threadIdx.x;
    const long long s = (long long)gridDim.x * blockDim.x;
    for (; idx < total; idx += s) {
        const int i  = (int)(idx >> 4);
        const int c4 = (int)(idx & 15) << 2;
        const float di = dinv[i];
        const float d2 = di * di;
        const float4 v = *(const float4*)(hw + (long long)i * 64 + c4);
        const float4 b = *(const float4*)(convb + c4);
        float4 o;
        o.x = v.x * d2 + b.x; o.y = v.y * d2 + b.y;
        o.z = v.z * d2 + b.z; o.w = v.w * d2 + b.w;
        *(float4*)(agg + (long long)i * 64 + c4) = o;
    }
}

// agg[dst] += hw[src] * dinv[src]*dinv[dst] ; thread per (edge, 4 channels)
__global__ void gcn_edge_scatter(const int* __restrict__ src,
                                 const int* __restrict__ dst,
                                 const float* __restrict__ dinv,
                                 const float* __restrict__ hw,
                                 float* __restrict__ agg, long long total)
{
    long long idx = (long long)blockIdx.x * blockDim.x + threadIdx.x;
    const long long s = (long long)gridDim.x * blockDim.x;
    for (; idx < total; idx += s) {
        const int e  = (int)(idx >> 4);
        const int c4 = (int)(idx & 15) << 2;
        const int sn = src[e];
        const int dn = dst[e];
        const float w = dinv[sn] * dinv[dn];
        const float4 v = *(const float4*)(hw + (long long)sn * 64 + c4);
        float* base = agg + (long long)dn * 64 + c4;
        atomicAdd(base + 0, v.x * w);
        atomicAdd(base + 1, v.y * w);
        atomicAdd(base + 2, v.z * w);
        atomicAdd(base + 3, v.w * w);
    }
}

// Per-channel sum / sum-of-squares: LDS partials -> global atomics.
// blockDim.x = 256 (multiple of 16) so each thread owns 4 fixed channels.
__global__ __launch_bounds__(256) void gcn_bn_stats(
    const float* __restrict__ agg, float* __restrict__ stats, long long total)
{
    __shared__ float ssum[64];
    __shared__ float ssq[64];
    const int t = threadIdx.x;
    if (t < 64) { ssum[t] = 0.0f; ssq[t] = 0.0f; }
    __syncthreads();

    long long idx = (long long)blockIdx.x * blockDim.x + t;   // over N*16
    const long long s = (long long)gridDim.x * blockDim.x;    // multiple of 16
    float4 lsum = {0, 0, 0, 0}, lsq = {0, 0, 0, 0};
    for (; idx < total; idx += s) {
        const float4 v = *(const float4*)(agg + (idx << 2));
        lsum.x += v.x; lsum.y += v.y; lsum.z += v.z; lsum.w += v.w;
        lsq.x += v.x * v.x; lsq.y += v.y * v.y;
        lsq.z += v.z * v.z; lsq.w += v.w * v.w;
    }
    const int c4 = (t & 15) << 2;
    atomicAdd(&ssum[c4 + 0], lsum.x); atomicAdd(&ssq[c4 + 0], lsq.x);
    atomicAdd(&ssum[c4 + 1], lsum.y); atomicAdd(&ssq[c4 + 1], lsq.y);
    atomicAdd(&ssum[c4 + 2], lsum.z); atomicAdd(&ssq[c4 + 2], lsq.z);
    atomicAdd(&ssum[c4 + 3], lsum.w); atomicAdd(&ssq[c4 + 3], lsq.w);
    __syncthreads();
    if (t < 64) {
        atomicAdd(&stats[t], ssum[t]);
        atomicAdd(&stats[64 + t], ssq[t]);
    }
}

// h = relu((agg - mean)*rsqrt(var+eps)*gamma + beta) + h ; total = N*16
__global__ void gcn_bn_apply(const float* __restrict__ agg,
                             const float* __restrict__ stats,
                             const float* __restrict__ gamma,
                             const float* __restrict__ beta,
                             float* __restrict__ h,
                             long long total, float invN)
{
    long long idx = (long long)blockIdx.x * blockDim.x + threadIdx.x;
    const long long s = (long long)gridDim.x * blockDim.x;
    for (; idx < total; idx += s) {
        const int c4 = (int)(idx & 15) << 2;
        const float4 sm = *(const float4*)(stats + c4);
        const float4 sq = *(const float4*)(stats + 64 + c4);
        const float4 gm = *(const float4*)(gamma + c4);
        const float4 bt = *(const float4*)(beta + c4);
        const float4 av = *(const float4*)(agg + (idx << 2));
        float4 hv = *(float4*)(h + (idx << 2));

        float mean, var, sc, sh, v;
        mean = sm.x * invN; var = sq.x * invN - mean * mean;
        sc = gm.x * rsqrtf(var + GCN_EPS); sh = bt.x - mean * sc;
        v = av.x * sc + sh; hv.x += v > 0.0f ? v : 0.0f;
        mean = sm.y * invN; var = sq.y * invN - mean * mean;
        sc = gm.y * rsqrtf(var + GCN_EPS); sh = bt.y - mean * sc;
        v = av.y * sc + sh; hv.y += v > 0.0f ? v : 0.0f;
        mean = sm.z * invN; var = sq.z * invN - mean * mean;
        sc = gm.z * rsqrtf(var + GCN_EPS); sh = bt.z - mean * sc;
        v = av.z * sc + sh; hv.z += v > 0.0f ? v : 0.0f;
        mean = sm.w * invN; var = sq.w * invN - mean * mean;
        sc = gm.w * rsqrtf(var + GCN_EPS); sh = bt.w - mean * sc;
        v = av.w * sc + sh; hv.w += v > 0.0f ? v : 0.0f;

        *(float4*)(h + (idx << 2)) = hv;
    }
}

// pooled[batch[i]] += h[i] ; total = N*16 (4 channels per thread)
__global__ void gcn_pool_accum(const float* __restrict__ h,
                               const int* __restrict__ batch,
                               float* __restrict__ pooled, long long total)
{
    long long idx = (long long)blockIdx.x * blockDim.x + threadIdx.x;
    const long long s = (long long)gridDim.x * blockDim.x;
    for (; idx < total; idx += s) {
        const int i  = (int)(idx >> 4);
        const int c4 = (int)(idx & 15) << 2;
        const float4 v = *(const float4*)(h + (long long)i * 64 + c4);
        float* base = pooled + (long long)batch[i] * 64 + c4;
        atomicAdd(base + 0, v.x);
        atomicAdd(base + 1, v.y);
        atomicAdd(base + 2, v.z);
        atomicAdd(base + 3, v.w);
    }
}

__global__ void gcn_pool_div(float* __restrict__ pooled,
                             const float* __restrict__ cnt, int total)
{
    const int i = blockIdx.x * blockDim.x + threadIdx.x;
    if (i < total) pooled[i] /= fmaxf(cnt[i >> 6], 1.0f);
}

// ---------------------------------------------------------------------------
// Host launch
// ---------------------------------------------------------------------------
static inline size_t gcn_align(size_t x) { return (x + 255) & ~(size_t)255; }

extern "C" void kernel_launch(void* const* d_in, const int* in_sizes, int n_in,
                              void* d_out, int out_size, void* d_ws, size_t ws_size,
                              hipStream_t stream)
{
    (void)in_sizes; (void)n_in; (void)out_size; (void)ws_size;

    const float* x      = (const float*)d_in[0];
    const int*   eidx   = (const int*)d_in[1];
    const int*   src    = eidx;            // edge_index[0]
    const int*   dst    = eidx + GCN_E;    // edge_index[1]
    const int*   batch  = (const int*)d_in[2];
    const float* W_in   = (const float*)d_in[3];
    const float* b_in   = (const float*)d_in[4];
    const float* convW  = (const float*)d_in[5];
    const float* convb  = (const float*)d_in[6];
    const float* gamma  = (const float*)d_in[7];
    const float* beta   = (const float*)d_in[8];
    const float* fcW    = (const float*)d_in[9];
    const float* fcb    = (const float*)d_in[10];
    float*       out    = (float*)d_out;

    // Workspace partition (~77 MB)
    char* w = (char*)d_ws;
    float* dinv   = (float*)w; w += gcn_align((size_t)GCN_N * 4);
    float* h      = (float*)w; w += gcn_align((size_t)GCN_N * 64 * 4);
    float* hw     = (float*)w; w += gcn_align((size_t)GCN_N * 64 * 4);
    float* agg    = (float*)w; w += gcn_align((size_t)GCN_N * 64 * 4);
    float* stats  = (float*)w; w += gcn_align(128 * 4);
    float* pooled = (float*)w; w += gcn_align((size_t)GCN_G * 64 * 4);
    float* cnt    = (float*)w; w += gcn_align((size_t)GCN_G * 4);

    const long long nodeVec = (long long)GCN_N * 16;   // float4 elements
    const long long edgeVec = (long long)GCN_E * 16;   // float4 elements

    const dim3 blk256(256);
    const dim3 gElem(2048);     // grid-stride for elementwise kernels
    const dim3 gScat(8192);     // grid-stride for edge scatter

    // ---- degree + dinv (self-loops => deg starts at 1) ----
    gcn_fill<<<dim3((GCN_N + 255) / 256), blk256, 0, stream>>>(dinv, GCN_N, 1.0f);
    gcn_count_accum<<<dim3((GCN_E + 255) / 256), blk256, 0, stream>>>(dst, dinv, GCN_E);
    gcn_rsqrt_inplace<<<dim3((GCN_N + 255) / 256), blk256, 0, stream>>>(dinv, GCN_N);

    // ---- input projection: h = x @ W_in + b_in ----
    gcn_gemm64_wmma<GCN_H, 5><<<dim3(GCN_N / 80, GCN_H / 16), dim3(32), 0, stream>>>(
        x, W_in, b_in, h, 0);

    // ---- GCN layers ----
    const float invN = 1.0f / (float)GCN_N;
    for (int l = 0; l < GCN_L; ++l) {
        const float* Wl = convW + (size_t)l * GCN_H * GCN_H;
        const float* bl = convb + (size_t)l * GCN_H;
        const float* gl = gamma + (size_t)l * GCN_H;
        const float* be = beta  + (size_t)l * GCN_H;

        gcn_gemm64_wmma<GCN_H, 5><<<dim3(GCN_N / 80, GCN_H / 16), dim3(32), 0, stream>>>(
            h, Wl, nullptr, hw, 0);

        gcn_agg_init<<<gElem, blk256, 0, stream>>>(hw, dinv, bl, agg, nodeVec);
        gcn_edge_scatter<<<gScat, blk256, 0, stream>>>(src, dst, dinv, hw, agg, edgeVec);

        gcn_fill<<<dim3(1), dim3(128), 0, stream>>>(stats, 128, 0.0f);
        gcn_bn_stats<<<dim3(1024), blk256, 0, stream>>>(agg, stats, nodeVec);
        gcn_bn_apply<<<gElem, blk256, 0, stream>>>(agg, stats, gl, be, h, nodeVec, invN);
    }

    // ---- global mean pool ----
    gcn_fill<<<dim3((GCN_G * 64 + 255) / 256), blk256, 0, stream>>>(pooled, GCN_G * 64, 0.0f);
    gcn_fill<<<dim3((GCN_G + 255) / 256), blk256, 0, stream>>>(cnt, GCN_G, 0.0f);
    gcn_count_accum<<<dim3((GCN_N + 255) / 256), blk256, 0, stream>>>(batch, cnt, GCN_N);
    gcn_pool_accum<<<gElem, blk256, 0, stream>>>(h, batch, pooled, nodeVec);
    gcn_pool_div<<<dim3((GCN_G * 64 + 255) / 256), blk256, 0, stream>>>(pooled, cnt, GCN_G * 64);

    // ---- final FC + sigmoid: out = sigmoid(pooled @ fcW + fcb), [512,16] ----
    gcn_gemm64_wmma<GCN_COUT, 4><<<dim3(GCN_G / 64, GCN_COUT / 16), dim3(32), 0, stream>>>(
        pooled, fcW, fcb, out, 1);
}